// DAGConstraintLayer_27290222198785
// MI455X (gfx1250) — compile-verified
//
#include <hip/hip_runtime.h>

// ---------------------------------------------------------------------------
// DAG constraint projection, MI455X (gfx1250), wave32.
//   - one lane per row; 32 q + 32 p values live in VGPRs, graph fully unrolled
//   - input staged global->LDS with gfx1250 async loads (coalesced 16B chunks),
//     output staged LDS->global with async stores; ASYNCcnt + s_wait_asynccnt
//   - LDS row stride padded to 144B to limit ds_load_b128 bank conflicts
// ---------------------------------------------------------------------------

#define N_NODES 32
#define ROWS_PER_BLOCK 256
#define ROW_STRIDE_F 36                      // 36 floats = 144 bytes (16B aligned)
#define N_ITERS 10
#define PROJ_TOL 1e-6f

typedef int v4i __attribute__((vector_size(16)));
typedef __attribute__((address_space(1))) v4i* gptr_v4i;
typedef __attribute__((address_space(3))) v4i* lptr_v4i;

// ---- gfx1250 async global<->LDS helpers (builtin if present, asm fallback) --

__device__ __forceinline__ void async_load_b128(const void* gsrc, void* ldst) {
#if __has_builtin(__builtin_amdgcn_global_load_async_to_lds_b128)
  __builtin_amdgcn_global_load_async_to_lds_b128(
      (gptr_v4i)gsrc, (lptr_v4i)ldst, 0, 0);
#else
  unsigned lds_addr = (unsigned)(unsigned long long)ldst;
  asm volatile("global_load_async_to_lds_b128 %0, %1, off"
               :: "v"(lds_addr), "v"(gsrc) : "memory");
#endif
}

__device__ __forceinline__ void async_store_b128(void* gdst, const void* lsrc) {
#if __has_builtin(__builtin_amdgcn_global_store_async_from_lds_b128)
  __builtin_amdgcn_global_store_async_from_lds_b128(
      (gptr_v4i)gdst, (lptr_v4i)lsrc, 0, 0);
#else
  unsigned lds_addr = (unsigned)(unsigned long long)lsrc;
  asm volatile("global_store_async_from_lds_b128 %0, %1, off"
               :: "v"(gdst), "v"(lds_addr) : "memory");
#endif
}

__device__ __forceinline__ void wait_async0() {
#if __has_builtin(__builtin_amdgcn_s_wait_asynccnt)
  __builtin_amdgcn_s_wait_asynccnt(0);
#else
  asm volatile("s_wait_asynccnt 0x0" ::: "memory");
#endif
}

__device__ __forceinline__ float fast_sigmoid(float v) {
  // sigmoid(v) = 1 / (1 + 2^(-v*log2(e)))  -> v_exp_f32 + v_rcp_f32
  return __builtin_amdgcn_rcpf(1.0f +
         __builtin_amdgcn_exp2f(v * -1.4426950408889634f));
}

// ---- projection on the fixed DAG: children(u) = {u+1, u+2} ∩ [0, 32) -------

__device__ __forceinline__ void project_row(const float* __restrict__ p,
                                            float* __restrict__ q) {
  for (int it = 0; it < N_ITERS; ++it) {
    // forward pass (topo order): scatter-min q[u] into children
#pragma unroll
    for (int u = 0; u < N_NODES - 1; ++u) {
      q[u + 1] = fminf(q[u + 1], q[u]);
      if (u + 2 < N_NODES) q[u + 2] = fminf(q[u + 2], q[u]);
    }
    // backward pass (reverse topo): q[u] = min(max(p[u], maxChild), minParent)
    float diff = 0.0f;
#pragma unroll
    for (int u = N_NODES - 1; u >= 0; --u) {
      float max_child;
      if (u == N_NODES - 1)      max_child = 0.0f;
      else if (u == N_NODES - 2) max_child = q[N_NODES - 1];
      else                       max_child = fmaxf(q[u + 1], q[u + 2]);
      float min_parent;
      if (u == 0)      min_parent = 1.0f;
      else if (u == 1) min_parent = q[0];
      else             min_parent = fminf(q[u - 1], q[u - 2]);
      float nv = fminf(fmaxf(p[u], max_child), min_parent);
      diff = fmaxf(diff, fabsf(nv - q[u]));
      q[u] = nv;
    }
    if (diff < PROJ_TOL) break;   // per-lane convergence (EXEC-masked)
  }
}

// ---------------------------------------------------------------------------

__global__ __launch_bounds__(ROWS_PER_BLOCK)
void dag_project_kernel(const float* __restrict__ x,
                        float* __restrict__ out,
                        int nrows) {
  __shared__ float tile[ROWS_PER_BLOCK * ROW_STRIDE_F];   // 36 KB

  const int tid = threadIdx.x;
  const long long rowBase = (long long)blockIdx.x * ROWS_PER_BLOCK;
  const bool fullTile = (rowBase + ROWS_PER_BLOCK) <= (long long)nrows;
  const bool active = fullTile || (rowBase + tid) < (long long)nrows;

  float p[N_NODES];
  float q[N_NODES];

  // -------- load + sigmoid --------
  if (fullTile) {
    // stage 32KB tile into LDS: chunk c = 16B at tileBase + c*16 (coalesced);
    // chunk c belongs to row r = c>>3, 16B-slot w = c&7 -> padded LDS layout
    const char* gsrc = (const char*)(x + rowBase * N_NODES);
#pragma unroll
    for (int k = 0; k < 8; ++k) {
      int c = tid + k * ROWS_PER_BLOCK;
      int r = c >> 3;
      int w = c & 7;
      async_load_b128(gsrc + (size_t)c * 16, &tile[r * ROW_STRIDE_F + w * 4]);
    }
    wait_async0();
    __syncthreads();

    const float4* rowp = (const float4*)&tile[tid * ROW_STRIDE_F];
#pragma unroll
    for (int k = 0; k < 8; ++k) {
      float4 v = rowp[k];
      p[4 * k + 0] = fast_sigmoid(v.x);
      p[4 * k + 1] = fast_sigmoid(v.y);
      p[4 * k + 2] = fast_sigmoid(v.z);
      p[4 * k + 3] = fast_sigmoid(v.w);
    }
  } else if (active) {
    const float4* g = (const float4*)(x + (rowBase + tid) * N_NODES);
#pragma unroll
    for (int k = 0; k < 8; ++k) {
      float4 v = g[k];
      p[4 * k + 0] = fast_sigmoid(v.x);
      p[4 * k + 1] = fast_sigmoid(v.y);
      p[4 * k + 2] = fast_sigmoid(v.z);
      p[4 * k + 3] = fast_sigmoid(v.w);
    }
  }

  // -------- iterative projection, all in VGPRs --------
  if (active) {
#pragma unroll
    for (int i = 0; i < N_NODES; ++i) q[i] = p[i];
    project_row(p, q);
  }

  // -------- store --------
  if (fullTile) {
    float4* rowp = (float4*)&tile[tid * ROW_STRIDE_F];
#pragma unroll
    for (int k = 0; k < 8; ++k)
      rowp[k] = make_float4(q[4 * k + 0], q[4 * k + 1],
                            q[4 * k + 2], q[4 * k + 3]);
    __syncthreads();

    char* gdst = (char*)(out + rowBase * N_NODES);
#pragma unroll
    for (int k = 0; k < 8; ++k) {
      int c = tid + k * ROWS_PER_BLOCK;
      int r = c >> 3;
      int w = c & 7;
      async_store_b128(gdst + (size_t)c * 16, &tile[r * ROW_STRIDE_F + w * 4]);
    }
    wait_async0();
  } else if (active) {
    float4* g = (float4*)(out + (rowBase + tid) * N_NODES);
#pragma unroll
    for (int k = 0; k < 8; ++k)
      g[k] = make_float4(q[4 * k + 0], q[4 * k + 1],
                         q[4 * k + 2], q[4 * k + 3]);
  }
}

// ---------------------------------------------------------------------------

extern "C" void kernel_launch(void* const* d_in, const int* in_sizes, int n_in,
                              void* d_out, int out_size, void* d_ws, size_t ws_size,
                              hipStream_t stream) {
  (void)n_in; (void)out_size; (void)d_ws; (void)ws_size;
  const float* x = (const float*)d_in[0];   // (B, 32) float32
  float* out = (float*)d_out;               // (B, 32) float32
  int nrows = in_sizes[0] / N_NODES;        // B
  if (nrows <= 0) return;
  int blocks = (nrows + ROWS_PER_BLOCK - 1) / ROWS_PER_BLOCK;
  dag_project_kernel<<<blocks, ROWS_PER_BLOCK, 0, stream>>>(x, out, nrows);
}